// EquivSetConv_53137335386864
// MI455X (gfx1250) — compile-verified
//
#include <hip/hip_runtime.h>

typedef __attribute__((ext_vector_type(2))) float v2f;
typedef __attribute__((ext_vector_type(8))) float v8f;

#define N_NODES_C 100000
#define N_EDGES_C 25000
#define NNZ_C     800000
#define D_C       64
#define ALPHA_C   0.5f

// Fragment-layout strides chosen so a full wave's fragment load covers all 64
// LDS banks exactly once:
//  - A: At[p][rr] (v2f). p-stride 144 v2f-rows = 1152B = 32 banks (mod 64), so
//    the hi=1 half-wave (p+1) is 32 banks from hi=0; lanes stride rr (2 banks).
//  - B: Bf[p][col] (v2f). p-stride 160 floats = 640B = 32 banks (mod 64).
#define A_PSTRIDE 144
#define B_PSTRIDE 160

__device__ __forceinline__ v8f wmma_f32(v2f a, v2f b, v8f c) {
  // D = A(16x4,f32) * B(4x16,f32) + C(16x16,f32)
  return __builtin_amdgcn_wmma_f32_16x16x4_f32(
      /*neg_a=*/false, a, /*neg_b=*/false, b,
      /*c_mod=*/(short)0, c, /*reuse_a=*/false, /*reuse_b=*/false);
}

// ---------------------------------------------------------------------------
// K1: Xd[r] += v * X[c]  (COO SpMM, 16 threads per nnz, float4 per thread)
// ---------------------------------------------------------------------------
__global__ void k_diffuse(const float* __restrict__ X,
                          const float* __restrict__ vals,
                          const int* __restrict__ rows,
                          const int* __restrict__ cols,
                          float* __restrict__ Xd) {
  int t  = blockIdx.x * blockDim.x + threadIdx.x;
  int i  = t >> 4;
  int c4 = (t & 15) << 2;
  __builtin_prefetch(vals + i + 1024, 0, 3);   // global_prefetch_b8: stream ahead
  __builtin_prefetch(cols + i + 1024, 0, 3);
  float v = vals[i];
  const float4 x = *reinterpret_cast<const float4*>(X + (size_t)cols[i] * D_C + c4);
  float* dst = Xd + (size_t)rows[i] * D_C + c4;
  atomicAdd(dst + 0, v * x.x);
  atomicAdd(dst + 1, v * x.y);
  atomicAdd(dst + 2, v * x.z);
  atomicAdd(dst + 3, v * x.w);
}

// ---------------------------------------------------------------------------
// K2/K5: Out[r] = (Ablend ? (1-alpha)*A[r]+alpha*Ablend[r] : A[r]) @ B + bias
// 256 threads = 8 waves; block covers 128 rows; each wave one 16x64 tile.
// fp32 WMMA 16x16x4; A and B staged in LDS in fragment layout (one
// ds_load_b64 per fragment, bank-conflict free).
// ---------------------------------------------------------------------------
__global__ void __launch_bounds__(256) k_gemm64(
    const float* __restrict__ A, const float* __restrict__ Ablend, float alpha,
    const float* __restrict__ B, const float* __restrict__ bias,
    float* __restrict__ Out, int nrows) {
  __shared__ float At[32 * A_PSTRIDE * 2];   // 36 KB: p=0..31, rr=0..127, v2f
  __shared__ float Bf[32 * B_PSTRIDE];       // 20 KB: p=0..31, col*2+{0,1}
  __shared__ float bs[64];

  const int tid  = threadIdx.x;
  const int row0 = blockIdx.x * 128;

  // stage A panel (with optional alpha-blend), zero-pad past nrows
  for (int idx = tid; idx < 128 * 16; idx += 256) {
    int rr = idx >> 4, c4 = idx & 15;
    int row = row0 + rr;
    float4 val = make_float4(0.f, 0.f, 0.f, 0.f);
    if (row < nrows) {
      val = *reinterpret_cast<const float4*>(A + (size_t)row * D_C + (c4 << 2));
      if (Ablend) {
        float4 w = *reinterpret_cast<const float4*>(Ablend + (size_t)row * D_C + (c4 << 2));
        val.x = (1.f - alpha) * val.x + alpha * w.x;
        val.y = (1.f - alpha) * val.y + alpha * w.y;
        val.z = (1.f - alpha) * val.z + alpha * w.z;
        val.w = (1.f - alpha) * val.w + alpha * w.w;
      }
    }
    int p = c4 << 1;                                   // k-pair index
    v2f lo; lo.x = val.x; lo.y = val.y;
    v2f hi2; hi2.x = val.z; hi2.y = val.w;
    *reinterpret_cast<v2f*>(&At[(p * A_PSTRIDE + rr) * 2])       = lo;
    *reinterpret_cast<v2f*>(&At[((p + 1) * A_PSTRIDE + rr) * 2]) = hi2;
  }
  // stage B (64x64) in fragment-interleaved layout
  for (int idx = tid; idx < 64 * 64; idx += 256) {
    int k = idx >> 6, col = idx & 63;
    Bf[(k >> 1) * B_PSTRIDE + (col << 1) + (k & 1)] = B[idx];
  }
  if (tid < 64) bs[tid] = bias[tid];
  __syncthreads();

  const int wave = tid >> 5, lane = tid & 31;
  const int lrow = lane & 15, hi = lane >> 4;
  const int rr   = wave * 16 + lrow;

  v8f acc[4] = {v8f{}, v8f{}, v8f{}, v8f{}};
#pragma unroll
  for (int k4 = 0; k4 < 16; ++k4) {
    const int p = (k4 << 1) + hi;
    v2f a = *reinterpret_cast<const v2f*>(&At[(p * A_PSTRIDE + rr) * 2]); // ds_load_b64
#pragma unroll
    for (int n = 0; n < 4; ++n) {
      v2f b = *reinterpret_cast<const v2f*>(
          &Bf[p * B_PSTRIDE + (((n << 4) + lrow) << 1)]);                 // ds_load_b64
      acc[n] = wmma_f32(a, b, acc[n]);
    }
  }

#pragma unroll
  for (int n = 0; n < 4; ++n)
#pragma unroll
    for (int j = 0; j < 8; ++j) {
      int row = row0 + wave * 16 + j + (hi << 3);      // C/D layout: VGPR j -> M=j(+8)
      if (row < nrows) {
        int col = (n << 4) + lrow;
        Out[(size_t)row * D_C + col] = acc[n][j] + bs[col];
      }
    }
}

// ---------------------------------------------------------------------------
// K3: Xe[edges[i]] += Y1[vertex[i]]
// ---------------------------------------------------------------------------
__global__ void k_scatter_edge(const float* __restrict__ Y1,
                               const int* __restrict__ vertex,
                               const int* __restrict__ edges,
                               float* __restrict__ Xe) {
  int t  = blockIdx.x * blockDim.x + threadIdx.x;
  int i  = t >> 4;
  int c4 = (t & 15) << 2;
  __builtin_prefetch(vertex + i + 1024, 0, 3);
  __builtin_prefetch(edges + i + 1024, 0, 3);
  const float4 y = *reinterpret_cast<const float4*>(Y1 + (size_t)vertex[i] * D_C + c4);
  float* dst = Xe + (size_t)edges[i] * D_C + c4;
  atomicAdd(dst + 0, y.x);
  atomicAdd(dst + 1, y.y);
  atomicAdd(dst + 2, y.z);
  atomicAdd(dst + 3, y.w);
}

// ---------------------------------------------------------------------------
// K4: per incidence pair p: row = [Xd[vertex[p]] | Xe[edges[p]]] (128 wide)
//     Xv[vertex[p]] += row @ W2 + b2
// Block = 256 threads = 8 waves = 128 pairs. LDS-staged gather in fragment
// layout, fp32 WMMA, atomic scatter of the 16x64 result per wave.
// ---------------------------------------------------------------------------
__global__ void __launch_bounds__(256) k_pair_gemm(
    const float* __restrict__ Xd, const float* __restrict__ Xe,
    const int* __restrict__ vertex, const int* __restrict__ edges,
    const float* __restrict__ W2, const float* __restrict__ b2,
    float* __restrict__ Xv) {
  __shared__ float At[64 * A_PSTRIDE * 2];   // 72 KB: p=0..63, rr=0..127, v2f
  __shared__ float Bf[64 * B_PSTRIDE];       // 40 KB: W2 fragment layout
  __shared__ float bs[64];

  const int tid = threadIdx.x;
  const int pb  = blockIdx.x * 128;

  __builtin_prefetch(vertex + pb + 128, 0, 3);  // next block's index stream
  __builtin_prefetch(edges + pb + 128, 0, 3);

  // gather 128 pairs: k 0..63 from Xd[vertex], k 64..127 from Xe[edges]
  for (int idx = tid; idx < 128 * 32; idx += 256) {
    int pp = idx >> 5, c4 = idx & 31;
    const float* src =
        (c4 < 16) ? (Xd + (size_t)vertex[pb + pp] * D_C + (c4 << 2))
                  : (Xe + (size_t)edges[pb + pp] * D_C + ((c4 - 16) << 2));
    float4 val = *reinterpret_cast<const float4*>(src);
    int p = c4 << 1;
    v2f lo; lo.x = val.x; lo.y = val.y;
    v2f hi2; hi2.x = val.z; hi2.y = val.w;
    *reinterpret_cast<v2f*>(&At[(p * A_PSTRIDE + pp) * 2])       = lo;
    *reinterpret_cast<v2f*>(&At[((p + 1) * A_PSTRIDE + pp) * 2]) = hi2;
  }
  // stage W2 (128x64) in fragment-interleaved layout
  for (int idx = tid; idx < 128 * 64; idx += 256) {
    int k = idx >> 6, col = idx & 63;
    Bf[(k >> 1) * B_PSTRIDE + (col << 1) + (k & 1)] = W2[idx];
  }
  if (tid < 64) bs[tid] = b2[tid];
  __syncthreads();

  const int wave = tid >> 5, lane = tid & 31;
  const int lrow = lane & 15, hi = lane >> 4;
  const int rr   = wave * 16 + lrow;

  v8f acc[4] = {v8f{}, v8f{}, v8f{}, v8f{}};
#pragma unroll
  for (int k4 = 0; k4 < 32; ++k4) {
    const int p = (k4 << 1) + hi;
    v2f a = *reinterpret_cast<const v2f*>(&At[(p * A_PSTRIDE + rr) * 2]); // ds_load_b64
#pragma unroll
    for (int n = 0; n < 4; ++n) {
      v2f b = *reinterpret_cast<const v2f*>(
          &Bf[p * B_PSTRIDE + (((n << 4) + lrow) << 1)]);                 // ds_load_b64
      acc[n] = wmma_f32(a, b, acc[n]);
    }
  }

  const int p0w = pb + wave * 16;
#pragma unroll
  for (int j = 0; j < 8; ++j) {
    int row = vertex[p0w + j + (hi << 3)];
#pragma unroll
    for (int n = 0; n < 4; ++n) {
      int col = (n << 4) + lrow;
      atomicAdd(&Xv[(size_t)row * D_C + col], acc[n][j] + bs[col]);
    }
  }
}

// ---------------------------------------------------------------------------
extern "C" void kernel_launch(void* const* d_in, const int* in_sizes, int n_in,
                              void* d_out, int out_size, void* d_ws, size_t ws_size,
                              hipStream_t stream) {
  const float* X      = (const float*)d_in[0];
  const float* W1_w   = (const float*)d_in[1];
  const float* W1_b   = (const float*)d_in[2];
  const float* W2_w   = (const float*)d_in[3];
  const float* W2_b   = (const float*)d_in[4];
  const float* W_w    = (const float*)d_in[5];
  const float* W_b    = (const float*)d_in[6];
  const float* dvals  = (const float*)d_in[7];
  const int*   vertex = (const int*)d_in[8];
  const int*   edges  = (const int*)d_in[9];
  const int*   drows  = (const int*)d_in[10];
  const int*   dcols  = (const int*)d_in[11];
  float* out = (float*)d_out;

  float* Xd = (float*)d_ws;
  float* Y1 = Xd + (size_t)N_NODES_C * D_C;
  float* Xe = Y1 + (size_t)N_NODES_C * D_C;
  float* Xv = Xe + (size_t)N_EDGES_C * D_C;

  hipMemsetAsync(Xd, 0, (size_t)N_NODES_C * D_C * sizeof(float), stream);
  hipMemsetAsync(Xe, 0, (size_t)N_EDGES_C * D_C * sizeof(float), stream);
  hipMemsetAsync(Xv, 0, (size_t)N_NODES_C * D_C * sizeof(float), stream);

  // 1. diffusion SpMM
  k_diffuse<<<NNZ_C * 16 / 256, 256, 0, stream>>>(X, dvals, drows, dcols, Xd);
  // 2. Y1 = Xd @ W1 + b1
  k_gemm64<<<(N_NODES_C + 127) / 128, 256, 0, stream>>>(
      Xd, nullptr, 0.f, W1_w, W1_b, Y1, N_NODES_C);
  // 3. Xe = segment_sum(Y1[vertex], edges)
  k_scatter_edge<<<NNZ_C * 16 / 256, 256, 0, stream>>>(Y1, vertex, edges, Xe);
  // 4. Xv = segment_sum([Xd[vertex]|Xe[edges]] @ W2 + b2, vertex)
  k_pair_gemm<<<NNZ_C / 128, 256, 0, stream>>>(Xd, Xe, vertex, edges, W2_w, W2_b, Xv);
  // 5. out = ((1-a)Xv + a*Xd) @ W + b
  k_gemm64<<<(N_NODES_C + 127) / 128, 256, 0, stream>>>(
      Xv, Xd, ALPHA_C, W_w, W_b, out, N_NODES_C);
}